// LabelSmoothing_48369921688197
// MI455X (gfx1250) — compile-verified
//
#include <hip/hip_runtime.h>
#include <cmath>

// LabelSmoothing KLDiv loss for gfx1250 (MI455X).
// Phase 1: bandwidth-bound row streaming (NT b128 loads + v_log_f32).
// Phase 2: tiny final reduction finished with V_WMMA_F32_16X16X4_F32.

typedef __attribute__((ext_vector_type(2))) float v2f;
typedef __attribute__((ext_vector_type(4))) float v4f;
typedef __attribute__((ext_vector_type(8))) float v8f;

#define BLOCK 256
#define SMOOTHING 0.1f
#define IGNORE_INDEX 0

// One block per row (B*L rows). Streams V floats, accumulates sum of log2(p),
// gathers log2(p[target]), emits per-row masked loss.
__global__ __launch_bounds__(BLOCK) void ls_row_kernel(
    const float* __restrict__ p, const int* __restrict__ tgt,
    float* __restrict__ row_loss, int V, float a, float b, float cst)
{
  const int    row  = blockIdx.x;
  const int    tid  = threadIdx.x;
  const size_t base = (size_t)row * (size_t)V;
  const v4f*   prow = reinterpret_cast<const v4f*>(p + base);
  const int    nvec = V >> 2;

  // 4 independent accumulators to pipeline the transcendental unit.
  float s0 = 0.f, s1 = 0.f, s2 = 0.f, s3 = 0.f;
  for (int i = tid; i < nvec; i += BLOCK) {
    v4f v = __builtin_nontemporal_load(prow + i);   // global_load_b128, TH=NT
    s0 += __builtin_amdgcn_logf(v.x);               // v_log_f32 (log2)
    s1 += __builtin_amdgcn_logf(v.y);
    s2 += __builtin_amdgcn_logf(v.z);
    s3 += __builtin_amdgcn_logf(v.w);
  }
  // Tail if V not a multiple of 4 (V=32000 is, but stay general).
  for (int i = (nvec << 2) + tid; i < V; i += BLOCK)
    s0 += __builtin_amdgcn_logf(p[base + (size_t)i]);

  float s = (s0 + s1) + (s2 + s3);   // per-thread partial, log2 domain

  // wave32 tree reduce
  #pragma unroll
  for (int off = 16; off > 0; off >>= 1)
    s += __shfl_down(s, off, 32);

  __shared__ float wsum[BLOCK / 32];
  if ((tid & 31) == 0) wsum[tid >> 5] = s;
  __syncthreads();

  if (tid == 0) {
    float t2 = 0.f;
    #pragma unroll
    for (int w = 0; w < BLOCK / 32; ++w) t2 += wsum[w];

    const float LN2 = 0.69314718055994530942f;
    const int   t   = tgt[row];
    const float logp_t   = __builtin_amdgcn_logf(p[base + (size_t)t]) * LN2;
    const float sum_logp = t2 * LN2;
    float loss = cst - a * (sum_logp - logp_t) - b * logp_t;
    row_loss[row] = (t == IGNORE_INDEX) ? 0.0f : loss;
  }
}

// Single-block reduction of n_rows partial losses. The last 32->1 step runs
// through V_WMMA_F32_16X16X4_F32 with an all-ones B matrix:
//   A 16x4 layout: lanes 0-15 supply A[m][0], lanes 16-31 supply A[m][2]
//   => D[m][n] = part_m + part_{m+16}; summing column N=0 (lane0 c0..7 holds
//   M=0..7, lane16 c0..7 holds M=8..15) gives the exact f32 total.
__global__ __launch_bounds__(BLOCK) void ls_reduce_kernel(
    const float* __restrict__ row_loss, const int* __restrict__ norm_ptr,
    float* __restrict__ out, int n_rows)
{
  const int tid = threadIdx.x;

  float s = 0.f;
  for (int i = tid; i < n_rows; i += BLOCK) s += row_loss[i];

  __shared__ float lds[BLOCK];
  lds[tid] = s;
  __syncthreads();

  if (tid < 32) {                       // wave 0 only: EXEC all-ones for WMMA
    float part = 0.f;
    #pragma unroll
    for (int k = 0; k < BLOCK / 32; ++k) part += lds[tid * (BLOCK / 32) + k];

    v2f a; a.x = part; a.y = 0.0f;      // A: 16x4 f32, zeros in unused K slots
    v2f b; b.x = 1.0f; b.y = 1.0f;      // B: 4x16 all ones
    v8f c = {};
    c = __builtin_amdgcn_wmma_f32_16x16x4_f32(
        /*neg_a=*/false, a, /*neg_b=*/false, b,
        /*c_mod=*/(short)0, c, /*reuse_a=*/false, /*reuse_b=*/false);

    float col = ((c[0] + c[1]) + (c[2] + c[3])) +
                ((c[4] + c[5]) + (c[6] + c[7]));          // column N=0 slice
    float other = __shfl(col, 16, 32);                    // lane16 -> lane0
    if (tid == 0) {
      float total = col + other;
      out[0] = total / (float)(*norm_ptr);                // low 32b of norm
    }
  }
}

extern "C" void kernel_launch(void* const* d_in, const int* in_sizes, int n_in,
                              void* d_out, int out_size, void* d_ws, size_t ws_size,
                              hipStream_t stream) {
  const float* p    = (const float*)d_in[0];
  const int*   tgt  = (const int*)d_in[1];
  const int*   norm = (const int*)d_in[2];

  const int rows = in_sizes[1];            // B*L = 8192
  const int V    = in_sizes[0] / rows;     // 32000

  // Smoothed-KL closed-form constants (host-side double precision).
  const double ad   = (double)SMOOTHING / (double)V;
  const double bd   = 1.0 - (double)SMOOTHING + ad;
  const double cstd = ad * std::log(ad) * (double)(V - 1) + bd * std::log(bd);

  float* row_loss = (float*)d_ws;          // rows * 4 bytes of scratch

  ls_row_kernel<<<rows, BLOCK, 0, stream>>>(p, tgt, row_loss, V,
                                            (float)ad, (float)bd, (float)cstd);
  ls_reduce_kernel<<<1, BLOCK, 0, stream>>>(row_loss, norm, (float*)d_out, rows);
}